// FrameTransformer_71081708749488
// MI455X (gfx1250) — compile-verified
//
#include <hip/hip_runtime.h>
#include <math.h>

// ---------------------------------------------------------------------------
// CDNA5 (gfx1250) wave32 WMMA helpers: D(16x16,f32) = A(16x4,f32)*B(4x16,f32)+C
// A layout: lane=(k/2)*16+m, vgpr=k%2  -> float2 per lane
// B layout: lane=(k/2)*16+n, vgpr=k%2  -> float2 per lane
// C/D layout: m = 8*(lane/16)+r, n = lane%16, r = vgpr 0..7
// ---------------------------------------------------------------------------
typedef float v2f __attribute__((ext_vector_type(2)));
typedef float v8f __attribute__((ext_vector_type(8)));

__device__ __forceinline__ v8f wmma4(v2f a, v2f b, v8f c) {
  return __builtin_amdgcn_wmma_f32_16x16x4_f32(
      /*neg_a=*/false, a, /*neg_b=*/false, b,
      /*c_mod=*/(short)0, c, /*reuse_a=*/false, /*reuse_b=*/false);
}

#define V8F_ZERO {0.f, 0.f, 0.f, 0.f, 0.f, 0.f, 0.f, 0.f}

// ---------------------------------------------------------------------------
// frame_norm: x (C,F,W), normalize over F per (c,w) column; scale/shift (C,F)
// ---------------------------------------------------------------------------
__global__ void frame_norm_k(const float* __restrict__ x, const float* __restrict__ g,
                             const float* __restrict__ b, float* __restrict__ out,
                             int C, int F, int W) {
  int w = blockIdx.x * blockDim.x + threadIdx.x;
  int c = blockIdx.y;
  if (w >= W) return;
  const float* xp = x + (size_t)c * F * W + w;
  float s = 0.f, ss = 0.f;
  for (int f = 0; f < F; ++f) { float v = xp[(size_t)f * W]; s += v; ss += v * v; }
  float mu = s / (float)F;
  float var = ss / (float)F - mu * mu;
  float rs = rsqrtf(var + 1e-5f);
  float* op = out + (size_t)c * F * W + w;
  for (int f = 0; f < F; ++f) {
    float v = xp[(size_t)f * W];
    op[(size_t)f * W] = (v - mu) * rs * g[c * F + f] + b[c * F + f];
  }
}

// ---------------------------------------------------------------------------
// Causal 3x3 conv as 9 tap-GEMMs on WMMA (K = CIN per tap, compile-time ->
// fully unrolled: loads clause up, WMMAs issue back-to-back).
//  in  : (CIN, Hin, W)  height pad (1,1), width pad (2,0) causal, stride (sh,1)
//  wgt : (Cout, CIN, 3, 3)    out : (Cout, Hout, W)
// Height guard is wave-uniform; width guard handled by clamp+select so every
// global load is unconditional. Optional fused SquaredReLU / 1x1 identity.
// ---------------------------------------------------------------------------
template <bool SQRELU, bool IDENT, int CIN, int CID>
__global__ void cconv3x3_wmma_k(const float* __restrict__ in, const float* __restrict__ wgt,
                                float* __restrict__ out,
                                int Hin, int Cout, int Hout, int W, int sh,
                                const float* __restrict__ xid, const float* __restrict__ idw) {
  const int lane = threadIdx.x;
  const int n = lane & 15, half = lane >> 4;
  const int w0  = blockIdx.x * 16;
  const int ho  = blockIdx.y;
  const int co0 = blockIdx.z * 16;
  const int hb  = ho * sh;
  const int aRow  = co0 + n;
  const bool aOk  = aRow < Cout;
  const int aRowC = aOk ? aRow : 0;                 // clamp, select later
  const size_t plane = (size_t)Hin * W;             // channel stride
  v8f acc = V8F_ZERO;

#pragma unroll
  for (int kh = 0; kh < 3; ++kh) {
    const int hi = hb + kh - 1;
    if (hi < 0 || hi >= Hin) continue;              // uniform scalar branch
#pragma unroll
    for (int kw = 0; kw < 3; ++kw) {
      const int wi  = w0 + n + kw - 2;              // causal left pad only
      const bool wOk = wi >= 0;                     // wi < W always
      const int wiC = wOk ? wi : 0;
      const float* wt = wgt + (kh * 3 + kw);        // tap slice, stride 9 over (co,ci)
      const float* bp = in + (size_t)hi * W + wiC;
#pragma unroll
      for (int c0 = 0; c0 < CIN; c0 += 4) {
        const int ka = c0 + 2 * half;
        const bool k0ok = ka < CIN;
        const bool k1ok = ka + 1 < CIN;
        const int ka0 = k0ok ? ka : 0;
        const int ka1 = k1ok ? ka + 1 : 0;
        v2f a, b;
        float a0 = wt[((size_t)aRowC * CIN + ka0) * 9];
        float a1 = wt[((size_t)aRowC * CIN + ka1) * 9];
        a.x = (aOk && k0ok) ? a0 : 0.f;
        a.y = (aOk && k1ok) ? a1 : 0.f;
        float b0 = bp[ka0 * plane];
        float b1 = bp[ka1 * plane];
        b.x = (k0ok && wOk) ? b0 : 0.f;
        b.y = (k1ok && wOk) ? b1 : 0.f;
        acc = wmma4(a, b, acc);
      }
    }
  }

  if (IDENT) {  // fused 1x1 identity conv, stride (sh,1), no padding
    const float* bp = xid + (size_t)hb * W + w0 + n;
#pragma unroll
    for (int c0 = 0; c0 < CID; c0 += 4) {
      const int ka = c0 + 2 * half;
      const bool k0ok = ka < CID;
      const bool k1ok = ka + 1 < CID;
      const int ka0 = k0ok ? ka : 0;
      const int ka1 = k1ok ? ka + 1 : 0;
      v2f a, b;
      float a0 = idw[(size_t)aRowC * CID + ka0];
      float a1 = idw[(size_t)aRowC * CID + ka1];
      a.x = (aOk && k0ok) ? a0 : 0.f;
      a.y = (aOk && k1ok) ? a1 : 0.f;
      float b0 = bp[ka0 * plane];
      float b1 = bp[ka1 * plane];
      b.x = k0ok ? b0 : 0.f;
      b.y = k1ok ? b1 : 0.f;
      acc = wmma4(a, b, acc);
    }
  }

#pragma unroll
  for (int r = 0; r < 8; ++r) {
    int co = co0 + 8 * half + r;
    if (co < Cout) {
      float v = acc[r];
      if (SQRELU) { v = fmaxf(v, 0.f); v = v * v; }
      out[((size_t)co * Hout + ho) * W + w0 + n] = v;
    }
  }
}

// ---------------------------------------------------------------------------
// mc_linear: out(c,G,W) = sum_f wgt(c,G,F) * x(c,F,W)
// FDIM compile-time -> fully unrolled K loop, float2 weight loads.
// G is a multiple of 16 for all call sites (32 or 128): no row guards.
// ---------------------------------------------------------------------------
template <int FDIM, bool SQRELU, bool RESID>
__global__ void mc_linear_wmma_k(const float* __restrict__ x, const float* __restrict__ wgt,
                                 float* __restrict__ out, int G, int W,
                                 const float* __restrict__ resid) {
  const int lane = threadIdx.x;
  const int n = lane & 15, half = lane >> 4;
  const int w0 = blockIdx.x * 16;
  const int g0 = blockIdx.y * 16;
  const int c  = blockIdx.z;
  const float* xp = x + (size_t)c * FDIM * W + w0 + n;
  const v2f* wrow = (const v2f*)(wgt + (size_t)c * G * FDIM + (size_t)(g0 + n) * FDIM);
  v8f acc = V8F_ZERO;

#pragma unroll
  for (int k0 = 0; k0 < FDIM; k0 += 4) {
    const int ka = k0 + 2 * half;
    v2f a = wrow[ka >> 1];                          // global_load_b64
    v2f b;
    b.x = xp[(size_t)(ka)     * W];
    b.y = xp[(size_t)(ka + 1) * W];
    acc = wmma4(a, b, acc);
  }

#pragma unroll
  for (int r = 0; r < 8; ++r) {
    int gg = g0 + 8 * half + r;
    float v = acc[r];
    if (SQRELU) { v = fmaxf(v, 0.f); v = v * v; }
    size_t oi = ((size_t)c * G + gg) * W + w0 + n;
    if (RESID) v += resid[oi];
    out[oi] = v;
  }
}

// ---------------------------------------------------------------------------
// Causal depthwise 1x7 conv (+ optional interleaved RoPE) and head re-layout:
//  y : (C, Ff, W)   dw : (C, 7)   o : (C, H, W, dh), f = head*dh + d
// ---------------------------------------------------------------------------
__global__ void dw7_rope_k(const float* __restrict__ y, const float* __restrict__ dw,
                           float* __restrict__ o, int C, int Ff, int W, int H, int dh,
                           int do_rope) {
  int idx = blockIdx.x * blockDim.x + threadIdx.x;
  int total = C * Ff * W;
  if (idx >= total) return;
  int w = idx % W;
  int f = (idx / W) % Ff;
  int c = idx / (W * Ff);
  const float* dwp = dw + c * 7;
  const float* yp = y + ((size_t)c * Ff + f) * W;
  float acc = 0.f;
#pragma unroll
  for (int t = 0; t < 7; ++t) { int wi = w - 6 + t; if (wi >= 0) acc += yp[wi] * dwp[t]; }
  float val = acc;
  int d = f % dh;
  if (do_rope) {
    const float* ypp = y + ((size_t)c * Ff + (f ^ 1)) * W;   // pair partner
    float accp = 0.f;
#pragma unroll
    for (int t = 0; t < 7; ++t) { int wi = w - 6 + t; if (wi >= 0) accp += ypp[wi] * dwp[t]; }
    int i = d >> 1;
    float theta = (float)w * powf(10000.f, -2.f * (float)i / (float)dh);
    float rot = (d & 1) ? accp : -accp;
    val = acc * cosf(theta) + rot * sinf(theta);
  }
  int head = f / dh;
  o[(((size_t)c * H + head) * W + w) * dh + d] = val;
}

// ---------------------------------------------------------------------------
// Attention per (c*H + head, q-tile of 16): one wave.
//  q,k,v : (CH, W, dh=8)   att : (C, H*dh, W)
// Entry : async-copy V slice (W*dh floats = 16 KB) to LDS with
//         global_load_async_to_lds_b128 (ASYNCcnt) — overlaps passes 1+softmax.
// Pass 1: S = scale*Q K^T -> LDS (16 x W, stride 516 -> conflict-free b64)
// Softmax rows with wave shuffles.
// Pass 2: s_wait_asynccnt 0, then O = P V with both operands from LDS.
// ---------------------------------------------------------------------------
#define PSTR 516
__global__ void attn_wmma_k(const float* __restrict__ q, const float* __restrict__ k,
                            const float* __restrict__ v, float* __restrict__ att,
                            int W, float scale) {
  __shared__ float P[16 * PSTR + 512 * 8];          // 33KB scores + 16KB V
  const int dh = 8;
  const int lane = threadIdx.x;
  const int n = lane & 15, half = lane >> 4;
  const int q0 = blockIdx.x * 16;
  const int ch = blockIdx.y;
  const float* qp = q + (size_t)ch * W * dh;
  const float* kp = k + (size_t)ch * W * dh;
  const float* vp = v + (size_t)ch * W * dh;
  float* VL = P + 16 * PSTR;                        // V in LDS, layout [w*dh + d]

  // Kick off async copy of V into LDS (ISA 10.7: ASYNCcnt-tracked DMA).
  // Low 32 bits of a generic pointer to __shared__ = LDS byte offset (ISA 10.2).
  {
    unsigned ldsb = (unsigned)(size_t)(VL) + (unsigned)(lane * 16);
    const float* g = vp + lane * 4;                 // 16B per lane per issue
    for (int it = 0; it < (512 * 8 * 4) / 512; ++it) {  // 32 issues x 512B
      asm volatile("global_load_async_to_lds_b128 %0, %1, off"
                   :: "v"(ldsb), "v"(g) : "memory");
      ldsb += 512;
      g += 128;
    }
  }

  // Pass 1: scores (overlaps with the async V copy)
  const v2f* q2 = (const v2f*)(qp + (size_t)(q0 + n) * dh);   // this lane's Q row
  for (int kt = 0; kt < W; kt += 16) {
    const v2f* k2 = (const v2f*)(kp + (size_t)(kt + n) * dh);
    v8f s = V8F_ZERO;
#pragma unroll
    for (int k0 = 0; k0 < 8; k0 += 4) {   // dh = 8 -> 2 WMMA steps
      v2f a = q2[(k0 >> 1) + half];
      v2f b = k2[(k0 >> 1) + half];
      s = wmma4(a, b, s);
    }
#pragma unroll
    for (int r = 0; r < 8; ++r)
      P[(8 * half + r) * PSTR + kt + n] = s[r] * scale;
  }
  __syncthreads();

  // Softmax over each of the 16 rows (length W), wave-cooperative
  for (int m = 0; m < 16; ++m) {
    float mx = -1e30f;
    for (int j = lane; j < W; j += 32) mx = fmaxf(mx, P[m * PSTR + j]);
#pragma unroll
    for (int off = 16; off > 0; off >>= 1) mx = fmaxf(mx, __shfl_xor(mx, off, 32));
    float sum = 0.f;
    for (int j = lane; j < W; j += 32) {
      float e = __expf(P[m * PSTR + j] - mx);
      P[m * PSTR + j] = e;
      sum += e;
    }
#pragma unroll
    for (int off = 16; off > 0; off >>= 1) sum += __shfl_xor(sum, off, 32);
    float inv = 1.f / sum;
    for (int j = lane; j < W; j += 32) P[m * PSTR + j] *= inv;
  }
  __syncthreads();

  // Wait for the async V copy before consuming it from LDS.
  asm volatile("s_wait_asynccnt 0x0" ::: "memory");

  // Pass 2: O = P V, both operands from LDS
  const bool vOk = n < dh;
  const int nC = vOk ? n : 0;                       // clamp (broadcast), select later
  v8f o = V8F_ZERO;
  for (int k0 = 0; k0 < W; k0 += 16) {
#pragma unroll
    for (int kk = 0; kk < 16; kk += 4) {            // 4 WMMA per outer iter
      const int ka = k0 + kk + 2 * half;
      v2f a = *(const v2f*)&P[n * PSTR + ka];       // ds_load_b64
      float b0 = VL[(ka)     * dh + nC];
      float b1 = VL[(ka + 1) * dh + nC];
      v2f b;
      b.x = vOk ? b0 : 0.f;
      b.y = vOk ? b1 : 0.f;
      o = wmma4(a, b, o);
    }
  }
  if (vOk) {
#pragma unroll
    for (int r = 0; r < 8; ++r) {
      int wq = q0 + 8 * half + r;
      att[((size_t)ch * dh + n) * W + wq] = o[r];
    }
  }
}

// ---------------------------------------------------------------------------
// Host orchestration
// ---------------------------------------------------------------------------
struct EncP { const float *nw, *nb, *c1, *c2, *id; };
struct LayP { const float *n1w, *n1b, *qpw, *qdw, *kpw, *kdw, *vpw, *vdw,
                          *outpw, *n2w, *n2b, *lin1, *lin2; };

extern "C" void kernel_launch(void* const* d_in, const int* in_sizes, int n_in,
                              void* d_out, int out_size, void* d_ws, size_t ws_size,
                              hipStream_t stream) {
  (void)out_size; (void)ws_size; (void)n_in;
  auto F = [&](int i) { return (const float*)d_in[i]; };

  const float* xin = nullptr;
  EncP enc[6]; LayP lay[2];
  // Detect flattening order: insertion order has x (1,048,576 elems) first;
  // jax-pytree sorted order puts enc_params first (stage0 'c1' = 36 elems), x last.
  bool insertion = (in_sizes[0] == 1 * 2 * 1024 * 512);
  if (insertion) {
    xin = F(0);
    for (int i = 0; i < 6; ++i) { int b = 1 + 5 * i;
      enc[i].nw = F(b + 0); enc[i].nb = F(b + 1);
      enc[i].c1 = F(b + 2); enc[i].c2 = F(b + 3); enc[i].id = F(b + 4);
    }
    for (int j = 0; j < 2; ++j) { int b = 31 + 13 * j;
      lay[j].n1w = F(b + 0);  lay[j].n1b = F(b + 1);
      lay[j].qpw = F(b + 2);  lay[j].qdw = F(b + 3);
      lay[j].kpw = F(b + 4);  lay[j].kdw = F(b + 5);
      lay[j].vpw = F(b + 6);  lay[j].vdw = F(b + 7);
      lay[j].outpw = F(b + 8);
      lay[j].n2w = F(b + 9);  lay[j].n2b = F(b + 10);
      lay[j].lin1 = F(b + 11); lay[j].lin2 = F(b + 12);
    }
  } else {  // sorted: enc[c1,c2,id,n_b,n_w]; lay[k_dw,k_pw,lin1,lin2,n1_b,n1_w,n2_b,n2_w,out_pw,q_dw,q_pw,v_dw,v_pw]; x last
    for (int i = 0; i < 6; ++i) { int b = 5 * i;
      enc[i].c1 = F(b + 0); enc[i].c2 = F(b + 1); enc[i].id = F(b + 2);
      enc[i].nb = F(b + 3); enc[i].nw = F(b + 4);
    }
    for (int j = 0; j < 2; ++j) { int b = 30 + 13 * j;
      lay[j].kdw = F(b + 0);  lay[j].kpw = F(b + 1);
      lay[j].lin1 = F(b + 2); lay[j].lin2 = F(b + 3);
      lay[j].n1b = F(b + 4);  lay[j].n1w = F(b + 5);
      lay[j].n2b = F(b + 6);  lay[j].n2w = F(b + 7);
      lay[j].outpw = F(b + 8);
      lay[j].qdw = F(b + 9);  lay[j].qpw = F(b + 10);
      lay[j].vdw = F(b + 11); lay[j].vpw = F(b + 12);
    }
    xin = F(56);
  }

  const int W = 512;
  static const int cins[6]  = {2, 2, 4, 8, 16, 32};
  static const int couts[6] = {2, 4, 8, 16, 32, 64};
  static const int feats[6] = {1024, 1024, 512, 256, 128, 64};
  static const int downs[6] = {0, 1, 1, 1, 1, 1};

  // Workspace layout (floats). Total footprint: 10 * 2^20 floats = 40 MB.
  float* ws = (float*)d_ws;
  const size_t M = (size_t)1 << 20;
  float* bufA = ws;           // 1M  stage output ping
  float* bufB = ws + 1 * M;   // 1M  stage output pong
  float* XN   = ws + 2 * M;   // 1M  frame_norm output
  float* ACT  = ws + 3 * M;   // 2M  conv1 activation
  float* H1   = ws + 3 * M;   // 4M  MLP hidden (reuses ACT..QR region)
  float* QY   = ws + 5 * M;   // 1M  projection temp
  float* QR   = ws + 6 * M;   // 1M  q (C,H,W,dh)
  float* KR   = ws + 7 * M;   // 1M  k
  float* VR   = ws + 8 * M;   // 1M  v
  float* ATT  = ws + 9 * M;   // 1M  attention output (C,F,W)

  // ---------------- Encoder ----------------
  const float* xc = xin;
  for (int i = 0; i < 6; ++i) {
    int Ci = cins[i], Co = couts[i], Hi = feats[i];
    int sh = downs[i] ? 2 : 1, Ho = Hi / sh;
    frame_norm_k<<<dim3((W + 255) / 256, Ci), 256, 0, stream>>>(
        xc, enc[i].nw, enc[i].nb, XN, Ci, Hi, W);
    float* xo = (i & 1) ? bufB : bufA;
    dim3 g1(W / 16, Hi, (Co + 15) / 16);
    dim3 g2(W / 16, Ho, (Co + 15) / 16);
#define LAUNCH_C1(CINV) \
    cconv3x3_wmma_k<true, false, CINV, 0><<<g1, 32, 0, stream>>>( \
        XN, enc[i].c1, ACT, Hi, Co, Hi, W, 1, nullptr, nullptr)
#define LAUNCH_C2(CINV, CIDV) \
    cconv3x3_wmma_k<false, true, CINV, CIDV><<<g2, 32, 0, stream>>>( \
        ACT, enc[i].c2, xo, Hi, Co, Ho, W, sh, xc, enc[i].id)
    switch (i) {
      case 0: LAUNCH_C1(2);  LAUNCH_C2(2, 2);   break;
      case 1: LAUNCH_C1(2);  LAUNCH_C2(4, 2);   break;
      case 2: LAUNCH_C1(4);  LAUNCH_C2(8, 4);   break;
      case 3: LAUNCH_C1(8);  LAUNCH_C2(16, 8);  break;
      case 4: LAUNCH_C1(16); LAUNCH_C2(32, 16); break;
      default: LAUNCH_C1(32); LAUNCH_C2(64, 32); break;
    }
#undef LAUNCH_C1
#undef LAUNCH_C2
    xc = xo;
  }
  float* xb = (float*)xc;  // bottleneck x: (64, 32, 512) in bufB

  // ---------------- Transformer layers ----------------
  const int C = 64, Fb = 32, Hh = 4, dh = 8, G1 = 128;
  const float scale = 0.17677669529663687f;  // 1/sqrt(h), h = 32
  for (int j = 0; j < 2; ++j) {
    const LayP& L = lay[j];
    frame_norm_k<<<dim3(2, C), 256, 0, stream>>>(xb, L.n1w, L.n1b, XN, C, Fb, W);
    // q / k / v projections + depthwise 1x7 + RoPE + head layout
    mc_linear_wmma_k<32, false, false><<<dim3(W / 16, Fb / 16, C), 32, 0, stream>>>(
        XN, L.qpw, QY, Fb, W, nullptr);
    dw7_rope_k<<<(C * Fb * W + 255) / 256, 256, 0, stream>>>(QY, L.qdw, QR, C, Fb, W, Hh, dh, 1);
    mc_linear_wmma_k<32, false, false><<<dim3(W / 16, Fb / 16, C), 32, 0, stream>>>(
        XN, L.kpw, QY, Fb, W, nullptr);
    dw7_rope_k<<<(C * Fb * W + 255) / 256, 256, 0, stream>>>(QY, L.kdw, KR, C, Fb, W, Hh, dh, 1);
    mc_linear_wmma_k<32, false, false><<<dim3(W / 16, Fb / 16, C), 32, 0, stream>>>(
        XN, L.vpw, QY, Fb, W, nullptr);
    dw7_rope_k<<<(C * Fb * W + 255) / 256, 256, 0, stream>>>(QY, L.vdw, VR, C, Fb, W, Hh, dh, 0);
    // attention (async V->LDS + WMMA QK^T + LDS softmax + WMMA PV)
    attn_wmma_k<<<dim3(W / 16, C * Hh), 32, 0, stream>>>(QR, KR, VR, ATT, W, scale);
    // out projection + residual: x = x + mc_linear(att, out_pw)
    mc_linear_wmma_k<32, false, true><<<dim3(W / 16, Fb / 16, C), 32, 0, stream>>>(
        ATT, L.outpw, xb, Fb, W, xb);
    // MLP: x = x + lin2(sqrelu(lin1(norm(x))))
    frame_norm_k<<<dim3(2, C), 256, 0, stream>>>(xb, L.n2w, L.n2b, XN, C, Fb, W);
    mc_linear_wmma_k<32, true, false><<<dim3(W / 16, G1 / 16, C), 32, 0, stream>>>(
        XN, L.lin1, H1, G1, W, nullptr);
    float* xo = (j == 1) ? (float*)d_out : xb;
    mc_linear_wmma_k<128, false, true><<<dim3(W / 16, Fb / 16, C), 32, 0, stream>>>(
        H1, L.lin2, xo, Fb, W, xb);
  }
}